// SDCIModel_730144441101
// MI455X (gfx1250) — compile-verified
//
#include <hip/hip_runtime.h>
#include <stdint.h>
#include <math.h>

typedef __attribute__((ext_vector_type(2))) float v2f;
typedef __attribute__((ext_vector_type(4))) float v4f;
typedef __attribute__((ext_vector_type(8))) float v8f;

#define B_BATCH 32
#define S_TOK   8192
#define D_FEA   128
#define K_TOP   4096
#define NCLUST  1024
#define CIN     512
#define HID     256
#define MEMN    128
#define NCLS    10

#define ASTRIDE 516            // padded LDS row stride (floats): 516%64==4 -> conflict-free ds_load_b64
#define MCHUNK  64             // clusters per block (64*4 = 256 token rows staged)
#define NCHUNKS (NCLUST / MCHUNK)   // 16

#define MASK_N     8388608u    // 32*1024*256 bernoulli elements
#define MASK_HALF  4194304u
#define MASK_WORDS (MASK_N / 32u)   // 262144 u32

// ---------------- threefry2x32 (JAX PRNG, key = jax.random.key(1234) -> [0,1234]) ----------------
__device__ __forceinline__ unsigned rotl32(unsigned x, int r) {
    return (x << r) | (x >> (32 - r));
}

// one threefry2x32 evaluation -> both 32-bit outputs (elements e and e+MASK_HALF)
__device__ __forceinline__ void threefry_pair(unsigned cx0, unsigned cx1,
                                              unsigned* y0, unsigned* y1) {
    const unsigned k0 = 0u, k1 = 1234u;
    const unsigned k2 = 0x1BD11BDAu ^ k0 ^ k1;
    unsigned x0 = cx0 + k0, x1 = cx1 + k1;
    const int r1[4] = {13, 15, 26, 6};
    const int r2[4] = {17, 29, 16, 24};
#pragma unroll
    for (int i = 0; i < 4; ++i) { x0 += x1; x1 = rotl32(x1, r1[i]); x1 ^= x0; }
    x0 += k1; x1 += k2 + 1u;
#pragma unroll
    for (int i = 0; i < 4; ++i) { x0 += x1; x1 = rotl32(x1, r2[i]); x1 ^= x0; }
    x0 += k2; x1 += k0 + 2u;
#pragma unroll
    for (int i = 0; i < 4; ++i) { x0 += x1; x1 = rotl32(x1, r1[i]); x1 ^= x0; }
    x0 += k0; x1 += k1 + 3u;
#pragma unroll
    for (int i = 0; i < 4; ++i) { x0 += x1; x1 = rotl32(x1, r2[i]); x1 ^= x0; }
    x0 += k1; x1 += k2 + 4u;
#pragma unroll
    for (int i = 0; i < 4; ++i) { x0 += x1; x1 = rotl32(x1, r1[i]); x1 ^= x0; }
    x0 += k2; x1 += k0 + 5u;
    *y0 = x0; *y1 = x1;
}

// ---------------- Kernel 0: bernoulli mask bits (p=0.5 -> mask = top bit clear) ----------------
// element e bit lives at maskbits[e>>5] bit (e&31); pair (e, e+MASK_HALF) from one eval.
__global__ void __launch_bounds__(256) mask_kernel(unsigned* __restrict__ maskbits) {
    unsigned tid = blockIdx.x * 256u + threadIdx.x;    // 0 .. MASK_HALF-1
    unsigned x0, x1;
    threefry_pair(tid, tid + MASK_HALF, &x0, &x1);
    // u < 0.5  <=>  ((bits>>9)|0x3f800000)-1.0f < 0.5f  <=>  bits>>31 == 0
    unsigned b0 = (~x0) >> 31;
    unsigned b1 = (~x1) >> 31;
    unsigned long long m0 = __ballot(b0);
    unsigned long long m1 = __ballot(b1);
    if ((threadIdx.x & 31) == 0) {
        unsigned w = tid >> 5;
        maskbits[w] = (unsigned)m0;
        maskbits[w + (MASK_HALF >> 5)] = (unsigned)m1;
    }
}

// ---------------- Kernel 1: per-token L2 norms (one wave32 per token) ----------------
__global__ void __launch_bounds__(256) norms_kernel(const float* __restrict__ x,
                                                    float* __restrict__ norms) {
    int wave = threadIdx.x >> 5;
    int lane = threadIdx.x & 31;
    int g = blockIdx.x * 8 + wave;                 // global token id, < 262144
    const float* p = x + (size_t)g * D_FEA + lane * 4;
    v4f v = *(const v4f*)p;
    float s = v.x * v.x + v.y * v.y + v.z * v.z + v.w * v.w;
#pragma unroll
    for (int off = 16; off > 0; off >>= 1) s += __shfl_xor(s, off, 32);
    if (lane == 0) norms[g] = sqrtf(s);
}

// ---------------- Kernel 2: per-batch full bitonic sort -> top-4096 indices (desc) ----------------
__global__ void __launch_bounds__(1024) topk_kernel(const float* __restrict__ norms,
                                                    unsigned* __restrict__ idx_out) {
    __shared__ unsigned long long keys[S_TOK];     // 64 KB LDS
    int b = blockIdx.x;
    int tid = threadIdx.x;
    for (int j = tid; j < S_TOK; j += 1024) {
        unsigned bits = __float_as_uint(norms[b * S_TOK + j]);
        // descending by norm; ties -> smaller index first (larger (8191-j) wins)
        keys[j] = ((unsigned long long)bits << 32) | (unsigned)(S_TOK - 1 - j);
    }
    for (unsigned size = 2; size <= S_TOK; size <<= 1) {
        for (unsigned stride = size >> 1; stride > 0; stride >>= 1) {
            __syncthreads();
            for (unsigned t = tid; t < (S_TOK / 2); t += 1024) {
                unsigned lo = ((t & ~(stride - 1)) << 1) | (t & (stride - 1));
                unsigned hi = lo + stride;
                bool asc = (lo & size) != 0;       // overall order: descending
                unsigned long long a = keys[lo], c = keys[hi];
                bool sw = asc ? (a > c) : (a < c);
                if (sw) { keys[lo] = c; keys[hi] = a; }
            }
        }
    }
    __syncthreads();
    for (int j = tid; j < K_TOP; j += 1024)
        idx_out[b * K_TOP + j] = (unsigned)(S_TOK - 1) - (unsigned)(keys[j] & 0xffffffffu);
}

// ---------------- Kernel 3: async-staged gather + f32 WMMA GEMM + mask + relu + column-sum ------
// grid = (16 M-chunks of 64 clusters, 32 batches); block = 256 = 8 waves.
// Stage: 256 gathered token rows -> LDS via global_load_async_to_lds_b128 (ASYNCcnt).
// Compute: wave owns 2 N-tiles, register-blocks 4 M-tiles (8 v8f accumulators).
__global__ void __launch_bounds__(256) gemm_kernel(const float* __restrict__ x,
                                                   const unsigned* __restrict__ idx,
                                                   const float* __restrict__ W_fc,
                                                   const float* __restrict__ b_fc,
                                                   const unsigned* __restrict__ maskbits,
                                                   float* __restrict__ xpart) {
    extern __shared__ float lds_a[];               // MCHUNK * ASTRIDE floats (132096 B dynamic)
    int chunk = blockIdx.x;                        // 0..15
    int b = blockIdx.y;                            // 0..31
    int wave = threadIdx.x >> 5;
    int lane = threadIdx.x & 31;
    int r = lane & 15;                             // A row / B col within a 16x16 tile
    int half = lane >> 4;                          // K-pair selector

    const unsigned* idxs = idx + b * K_TOP + chunk * (MCHUNK * 4);
    const float* xb = x + (size_t)b * S_TOK * D_FEA;

    // ---- stage gathered A tile into LDS with CDNA5 async loads (one b128 per lane per row) ----
    unsigned abase = (unsigned)(size_t)(&lds_a[0]);
    for (int i = 0; i < 32; ++i) {
        int rr = wave * 32 + i;                    // token row 0..255 within chunk
        unsigned tok = idxs[rr];
        const float* g = xb + (size_t)tok * D_FEA + lane * 4;
        unsigned loff = abase +
            (unsigned)((((rr >> 2) * ASTRIDE) + (rr & 3) * D_FEA + lane * 4) * 4);
        asm volatile("global_load_async_to_lds_b128 %0, %1, off"
                     :: "v"(loff), "v"(g) : "memory");
    }
    asm volatile("s_wait_asynccnt 0" ::: "memory");
    __syncthreads();

    // ---- register-blocked WMMA: 4 M-tiles x 2 N-tiles per wave ----
    int tn0 = wave * 2, tn1 = tn0 + 1;
    const float* pb0 = W_fc + (size_t)(tn0 * 16 + r) * CIN + 2 * half;  // B[k][n] = W_fc[n][k]
    const float* pb1 = pb0 + (size_t)16 * CIN;
    const float* pa  = lds_a + r * ASTRIDE + 2 * half;

    v8f c[4][2];
#pragma unroll
    for (int mt = 0; mt < 4; ++mt) {
        c[mt][0] = (v8f){0.f, 0.f, 0.f, 0.f, 0.f, 0.f, 0.f, 0.f};
        c[mt][1] = (v8f){0.f, 0.f, 0.f, 0.f, 0.f, 0.f, 0.f, 0.f};
    }

#pragma unroll 2
    for (int kk = 0; kk < 128; ++kk) {             // K step of 4 per WMMA
        int k4 = kk * 4;
        v2f b0 = *(const v2f*)(pb0 + k4);
        v2f b1 = *(const v2f*)(pb1 + k4);
#pragma unroll
        for (int mt = 0; mt < 4; ++mt) {
            v2f a = *(const v2f*)(pa + mt * 16 * ASTRIDE + k4);
            c[mt][0] = __builtin_amdgcn_wmma_f32_16x16x4_f32(
                           false, a, false, b0, (short)0, c[mt][0], false, false);
            c[mt][1] = __builtin_amdgcn_wmma_f32_16x16x4_f32(
                           false, a, false, b1, (short)0, c[mt][1], false, false);
        }
    }

    // ---- epilogue: bias -> mask bit -> relu -> column sums (wave-exclusive columns) ----
    // wave's 2 N-tiles span columns [wave*32, wave*32+32): both tiles' mask bits for a row
    // live in one word maskbits[row*8 + wave] at bits r (tn0) and 16+r (tn1).
    float bfc0 = b_fc[tn0 * 16 + r];
    float bfc1 = b_fc[tn1 * 16 + r];
    float acc0 = 0.f, acc1 = 0.f;
#pragma unroll
    for (int mt = 0; mt < 4; ++mt) {
        unsigned base_n = (unsigned)(b * NCLUST + chunk * MCHUNK + mt * 16);
#pragma unroll
        for (int v = 0; v < 8; ++v) {
            unsigned m = (unsigned)v + 8u * (unsigned)half;
            unsigned mw = maskbits[(base_n + m) * (unsigned)(HID / 32) + (unsigned)wave];
            float f0 = c[mt][0][v] + bfc0;
            f0 = ((mw >> r) & 1u) ? f0 : 0.f;
            acc0 += (f0 > 0.f) ? f0 : 0.f;
            float f1 = c[mt][1][v] + bfc1;
            f1 = ((mw >> (16 + r)) & 1u) ? f1 : 0.f;
            acc1 += (f1 > 0.f) ? f1 : 0.f;
        }
    }
    // fold the two row-halves (lane L and L+16 hold the same column)
    acc0 += __shfl_xor(acc0, 16, 32);
    acc1 += __shfl_xor(acc1, 16, 32);
    if (lane < 16) {
        float* xp = xpart + ((size_t)b * NCHUNKS + chunk) * HID;
        xp[tn0 * 16 + lane] = acc0;
        xp[tn1 * 16 + lane] = acc1;
    }
}

// ---------------- Kernel 4: chunk-partial reduce + mean + memory update + output head ----------------
__global__ void __launch_bounds__(256) head_kernel(const float* __restrict__ xpart,
                                                   const float* __restrict__ memory,
                                                   const float* __restrict__ W_mem,
                                                   const float* __restrict__ b_mem,
                                                   const float* __restrict__ W_out,
                                                   const float* __restrict__ b_out,
                                                   float* __restrict__ out) {
    __shared__ float xc[B_BATCH * HID];            // 32 KB
    __shared__ float um[B_BATCH * MEMN];           // 16 KB
    int t = threadIdx.x;
    const float inv = 1.0f / (float)NCLUST;
    for (int i = t; i < B_BATCH * HID; i += 256) { // fixed-order deterministic reduction
        int b = i >> 8, h = i & 255;
        float s = 0.f;
        for (int ch = 0; ch < NCHUNKS; ++ch)
            s += xpart[((size_t)b * NCHUNKS + ch) * HID + h];
        xc[i] = s * inv;
    }
    __syncthreads();
    for (int i = t; i < B_BATCH * MEMN; i += 256) {
        int b = i >> 7, m = i & 127;
        float s = b_mem[m] + memory[i];
        const float* wr = W_mem + (size_t)m * HID;
        const float* xcb = xc + b * HID;
        for (int hh = 0; hh < HID; ++hh) s += xcb[hh] * wr[hh];
        float rr = (s > 0.f) ? s : 0.f;
        um[i] = rr;
        out[B_BATCH * NCLS + i] = rr;              // updated_memory after output
    }
    __syncthreads();
    for (int i = t; i < B_BATCH * NCLS; i += 256) {
        int b = i / NCLS, o = i - b * NCLS;
        float s = b_out[o];
        const float* wr = W_out + (size_t)o * MEMN;
        const float* u = um + b * MEMN;
        for (int m = 0; m < MEMN; ++m) s += u[m] * wr[m];
        out[i] = s;
    }
}

extern "C" void kernel_launch(void* const* d_in, const int* in_sizes, int n_in,
                              void* d_out, int out_size, void* d_ws, size_t ws_size,
                              hipStream_t stream) {
    const float* x      = (const float*)d_in[0];
    const float* memory = (const float*)d_in[1];
    const float* W_fc   = (const float*)d_in[2];
    const float* b_fc   = (const float*)d_in[3];
    const float* W_mem  = (const float*)d_in[4];
    const float* b_mem  = (const float*)d_in[5];
    const float* W_out  = (const float*)d_in[6];
    const float* b_out  = (const float*)d_in[7];
    float* out = (float*)d_out;

    char* ws = (char*)d_ws;
    float*    norms    = (float*)ws;                                  // 1 MiB
    unsigned* idx      = (unsigned*)(ws + (size_t)1048576);           // 512 KiB
    float*    xpart    = (float*)(ws + (size_t)1048576 + 524288);     // 512 KiB
    unsigned* maskbits = (unsigned*)(ws + (size_t)1048576 + 524288 + 524288); // 1 MiB

    mask_kernel<<<MASK_HALF / 256, 256, 0, stream>>>(maskbits);
    norms_kernel<<<(B_BATCH * S_TOK) / 8, 256, 0, stream>>>(x, norms);
    topk_kernel<<<B_BATCH, 1024, 0, stream>>>(norms, idx);
    gemm_kernel<<<dim3(NCHUNKS, B_BATCH), 256, MCHUNK * ASTRIDE * 4, stream>>>(
        x, idx, W_fc, b_fc, maskbits, xpart);
    head_kernel<<<1, 256, 0, stream>>>(xpart, memory, W_mem, b_mem, W_out, b_out, out);
}